// GaussSoftmaxEtm_31233002176928
// MI455X (gfx1250) — compile-verified
//
#include <hip/hip_runtime.h>
#include <math.h>

// Problem constants (match reference)
#define V  32000
#define T  128
#define E  64
#define NN 1024

#define BOUND   1.41421356237f        // sqrt(2)
#define LSMIN  (-2.30258509299f)      // log(0.1)
#define LSMAX   (2.30258509299f)      // log(10)
#define CTERM   117.624132250f        // E * log(2*pi)

typedef float v2f __attribute__((ext_vector_type(2)));
typedef float v8f __attribute__((ext_vector_type(8)));

// monotone float<->uint map so atomicMax(uint) implements float max
__device__ __forceinline__ unsigned f2o(float f) {
    unsigned b = __float_as_uint(f);
    return (b & 0x80000000u) ? ~b : (b | 0x80000000u);
}
__device__ __forceinline__ float o2f(unsigned u) {
    unsigned b = (u & 0x80000000u) ? (u & 0x7FFFFFFFu) : ~u;
    return __uint_as_float(b);
}

// ---------------------------------------------------------------- init stats
__global__ void k_init(unsigned* __restrict__ maxbits, float* __restrict__ sumexp) {
    int i = threadIdx.x;
    if (i < T) { maxbits[i] = 0u; sumexp[i] = 0.0f; }
}

// ---------------------------------------------------------------- logits [V,T]
// grid (V/8, 2), block 256. blockIdx.y selects a 64-topic half (32KB LDS).
__global__ void k_logits(const float* __restrict__ mu, const float* __restrict__ ls,
                         const float* __restrict__ mu_c, const float* __restrict__ ls_c,
                         float* __restrict__ logits) {
    __shared__ float s_sc[E * 64];   // sigma_c transposed [e][t]  (32KB total w/ s_mc)
    __shared__ float s_mc[E * 64];   // mu_c    transposed [e][t]
    __shared__ float s_sv[4][E];     // 4 vocab rows: sigma
    __shared__ float s_mv[4][E];     // 4 vocab rows: mu

    const int tid = threadIdx.x;
    const int t0  = blockIdx.y * 64;

    // stage topic params (clamp + exp applied), transposed for conflict-free reads
    for (int i = tid; i < 64 * E; i += 256) {
        int tl = i >> 6, e = i & 63;
        int src = (t0 + tl) * E + e;
        float m = mu_c[src]; m = fminf(fmaxf(m, -BOUND), BOUND);
        float l = ls_c[src]; l = fminf(fmaxf(l, LSMIN), LSMAX);
        s_mc[e * 64 + tl] = m;
        s_sc[e * 64 + tl] = __expf(l);
    }

    const int tl = tid & 63;          // local topic
    const int vs = tid >> 6;          // 0..3 vocab sub-row
    const int v0 = blockIdx.x * 8;

    for (int vv = 0; vv < 8; vv += 4) {
        __syncthreads();              // covers topic staging (iter 0) / prev compute
        { // stage 4 vocab rows
            int r = tid >> 6, e = tid & 63;
            int row = (v0 + vv + r) * E + e;
            float m = mu[row]; s_mv[r][e] = fminf(fmaxf(m, -BOUND), BOUND);
            float l = ls[row]; s_sv[r][e] = __expf(fminf(fmaxf(l, LSMIN), LSMAX));
        }
        __syncthreads();

        float acc = 0.0f;
        #pragma unroll 8
        for (int e = 0; e < E; ++e) {
            float s = s_sv[vs][e] + s_sc[e * 64 + tl];
            float d = s_mv[vs][e] - s_mc[e * 64 + tl];
            acc += __logf(s) + d * d / s;
        }
        int v = v0 + vv + vs;
        logits[v * T + t0 + tl] = -0.5f * (acc + CTERM);
    }
}

// ---------------------------------------------------------------- per-topic max
// grid V/64, block 256: threads sweep topics (coalesced), 64 rows/block
__global__ void k_max(const float* __restrict__ logits, unsigned* __restrict__ maxbits) {
    __shared__ float red[256];
    int tid = threadIdx.x;
    int t = tid & 127, vh = tid >> 7;
    int v0 = blockIdx.x * 64;
    float m = -3.4e38f;
    for (int v = vh; v < 64; v += 2)
        m = fmaxf(m, logits[(v0 + v) * T + t]);
    red[tid] = m;
    __syncthreads();
    if (tid < 128)
        atomicMax(&maxbits[t], f2o(fmaxf(red[t], red[t + 128])));
}

// ---------------------------------------------------------------- per-topic sum(exp)
__global__ void k_sum(const float* __restrict__ logits, const unsigned* __restrict__ maxbits,
                      float* __restrict__ sumexp) {
    __shared__ float red[256];
    int tid = threadIdx.x;
    int t = tid & 127, vh = tid >> 7;
    int v0 = blockIdx.x * 64;
    float mx = o2f(maxbits[t]);
    float s = 0.0f;
    for (int v = vh; v < 64; v += 2)
        s += __expf(logits[(v0 + v) * T + t] - mx);
    red[tid] = s;
    __syncthreads();
    if (tid < 128)
        atomicAdd(&sumexp[t], red[t] + red[t + 128]);
}

// ---------------------------------------------------------------- W = softmax weights
__global__ void k_weights(const float* __restrict__ logits, const unsigned* __restrict__ maxbits,
                          const float* __restrict__ sumexp, float* __restrict__ Wm) {
    int i = blockIdx.x * 256 + threadIdx.x;   // grid = V*T/256
    int t = i & (T - 1);
    Wm[i] = __expf(logits[i] - o2f(maxbits[t])) / sumexp[t];
}

// ---------------------------------------------------------------- out = W @ x  (f32 WMMA)
// grid (NN/128, V/16), block 256 (8 waves). Wave w: 16x16 tile at (m0, nblk + 16w).
// x K-slab staged in LDS, pad 136 so the two K-halves of one ds_load hit disjoint banks.
#define KC   64
#define XPAD 136
__global__ void k_gemm(const float* __restrict__ Wm, const float* __restrict__ xs,
                       float* __restrict__ out) {
    __shared__ float s_x[KC * XPAD];          // ~34.8 KB

    const int tid  = threadIdx.x;
    const int wave = tid >> 5;
    const int lane = tid & 31;
    const int m0   = blockIdx.y * 16;
    const int nblk = blockIdx.x * 128;
    const int ncol = (wave << 4) + (lane & 15);         // B/C column within block slab
    const int mrow = m0 + (lane & 15);                  // A row (lane&15 = M)
    const int khalf = (lane >> 4) << 1;                 // upper half-wave owns K+2,K+3

    v8f c = {0.f, 0.f, 0.f, 0.f, 0.f, 0.f, 0.f, 0.f};

    for (int kc = 0; kc < T; kc += KC) {
        __syncthreads();
        for (int i = tid; i < KC * 128; i += 256) {     // stage x[kc..kc+KC) x 128 cols
            int r = i >> 7, col = i & 127;
            s_x[r * XPAD + col] = xs[(kc + r) * NN + nblk + col];
        }
        __syncthreads();

        #pragma unroll
        for (int k = 0; k < KC; k += 4) {
            int kb = k + khalf;
            // A 16x4 f32: lane&15=M; VGPR0=K(kb), VGPR1=K(kb+1); half-wave -> kb+=2
            v2f a = *(const v2f*)(Wm + mrow * T + kc + kb);   // 8B aligned (kb even)
            // B 4x16 f32: lane&15=N; VGPR0=row kb, VGPR1=row kb+1; half-wave -> +2
            v2f b;
            b.x = s_x[kb * XPAD + ncol];
            b.y = s_x[(kb + 1) * XPAD + ncol];
            c = __builtin_amdgcn_wmma_f32_16x16x4_f32(
                    /*neg_a=*/false, a, /*neg_b=*/false, b,
                    /*c_mod=*/(short)0, c, /*reuse_a=*/false, /*reuse_b=*/false);
        }
    }

    // C/D 16x16 f32 layout: VGPR r -> M=r (lanes 0-15) / M=r+8 (lanes 16-31), N=lane&15
    const int nout  = nblk + ncol;
    const int mbase = m0 + ((lane >> 4) << 3);
    #pragma unroll
    for (int r = 0; r < 8; ++r)
        out[(mbase + r) * NN + nout] = c[r];
}

// ---------------------------------------------------------------- launch
extern "C" void kernel_launch(void* const* d_in, const int* in_sizes, int n_in,
                              void* d_out, int out_size, void* d_ws, size_t ws_size,
                              hipStream_t stream) {
    const float* x    = (const float*)d_in[0];   // [T, NN]
    const float* mu   = (const float*)d_in[1];   // [V, E]
    const float* ls   = (const float*)d_in[2];   // [V, E]
    const float* mu_c = (const float*)d_in[3];   // [T, E]
    const float* ls_c = (const float*)d_in[4];   // [T, E]
    float* out = (float*)d_out;                  // [V, NN]

    float*    logits  = (float*)d_ws;                        // V*T floats
    float*    Wm      = logits + (size_t)V * T;              // V*T floats
    unsigned* maxbits = (unsigned*)(Wm + (size_t)V * T);     // T uints
    float*    sumexp  = (float*)(maxbits + T);               // T floats

    k_init   <<<1, 256, 0, stream>>>(maxbits, sumexp);
    k_logits <<<dim3(V / 8, 2), 256, 0, stream>>>(mu, ls, mu_c, ls_c, logits);
    k_max    <<<V / 64, 256, 0, stream>>>(logits, maxbits);
    k_sum    <<<V / 64, 256, 0, stream>>>(logits, maxbits, sumexp);
    k_weights<<<(V * T) / 256, 256, 0, stream>>>(logits, maxbits, sumexp, Wm);
    k_gemm   <<<dim3(NN / 128, V / 16), 256, 0, stream>>>(Wm, x, out);
    (void)in_sizes; (void)n_in; (void)out_size; (void)ws_size;
}